// EncoderBlock_8040178778231
// MI455X (gfx1250) — compile-verified
//
#include <hip/hip_runtime.h>
#include <math.h>

// ---------------------------------------------------------------------------
// LLaMA decoder layer, batch=32, seq=1, decode @ start_pos=1024.
// Bandwidth-bound (~950MB traffic, ~11 GFLOP): fp32 end-to-end, weights
// streamed once via V_WMMA_F32_16X16X4_F32 (M=32 -> two 16-row A tiles,
// 2 n-subtiles per wave sharing A). Deterministic split-K partials + reduce.
// ---------------------------------------------------------------------------

#define DIM     4096
#define NHQ     32
#define NHKV    8
#define HD      128
#define HIDDEN  11008
#define BATCH   32
#define MAXSEQ  2048
#define SPLITK  8

typedef __attribute__((ext_vector_type(2))) float v2f;
typedef __attribute__((ext_vector_type(8))) float v8f;

#define WMMA_F32(A, B, C) \
    __builtin_amdgcn_wmma_f32_16x16x4_f32(false, (A), false, (B), (short)0, (C), false, false)

// ------------------------------ RMSNorm ------------------------------------
__global__ __launch_bounds__(256) void rmsnorm_k(const float* __restrict__ x,
                                                 const float* __restrict__ w,
                                                 float* __restrict__ out) {
    __shared__ float red[256];
    __shared__ float sinv;
    const int b = blockIdx.x, tid = threadIdx.x;
    const float* row = x + (size_t)b * DIM;
    float ss = 0.f;
    for (int i = tid; i < DIM; i += 256) { float v = row[i]; ss += v * v; }
    red[tid] = ss;
    __syncthreads();
    for (int o = 128; o > 0; o >>= 1) {
        if (tid < o) red[tid] += red[tid + o];
        __syncthreads();
    }
    if (tid == 0) sinv = 1.f / (sqrtf(red[0] / (float)DIM) + 1e-5f);
    __syncthreads();
    const float inv = sinv;
    float* o = out + (size_t)b * DIM;
    for (int i = tid; i < DIM; i += 256) o[i] = w[i] * row[i] * inv;
}

// --------------------- 32 x K x N GEMM via fp32 WMMA -----------------------
// H: 32 x K (row major), W: K x N (row major), P: SPLITK x 32 x N partials.
// One wave owns a 32(m) x 32(n) tile (2 n-subtiles reuse each A fragment);
// block = 8 waves = 256 columns. Addressing is strength-reduced to pointer
// increments so the loop is loads + 4x WMMA, not 64-bit address math.
// A frag (16x4 f32): lane m = lane&15, vgpr j holds K = 2*(lane>>4)+j.
// B frag (4x16 f32): lane n = lane&15, vgpr j holds K = 2*(lane>>4)+j.
// C/D (16x16 f32): vgpr r <-> (m = r + 8*(lane>>4), n = lane&15).
__global__ __launch_bounds__(256) void gemm32_wmma(const float* __restrict__ H,
                                                   const float* __restrict__ W,
                                                   float* __restrict__ P,
                                                   int K, int N) {
    const int lane = threadIdx.x & 31;
    const int wave = threadIdx.x >> 5;
    const int lo   = lane & 15;
    const int hi   = lane >> 4;
    const int n0   = blockIdx.x * 256 + wave * 32;
    const int Kslice = K / SPLITK;
    const int kb0    = blockIdx.y * Kslice;
    const int nsteps = Kslice / 4;
    const size_t sN  = (size_t)N;

    const float* ap0 = H + (size_t)lo * K + kb0 + 2 * hi;        // rows 0..15
    const float* ap1 = H + (size_t)(lo + 16) * K + kb0 + 2 * hi; // rows 16..31
    const float* wp0 = W + (size_t)(kb0 + 2 * hi) * sN + n0 + lo;
    const float* wp1 = wp0 + 16;

    v8f c00 = {}, c01 = {}, c10 = {}, c11 = {};
#pragma unroll 4
    for (int it = 0; it < nsteps; ++it) {
        if ((it & 15) == 0 && (it + 16) < nsteps)   // uniform; stream ahead
            __builtin_prefetch(wp0 + 64 * sN, 0, 3);
        v2f a0 = *(const v2f*)ap0;
        v2f a1 = *(const v2f*)ap1;
        v2f b0, b1;
        b0.x = wp0[0];   b0.y = wp0[sN];
        b1.x = wp1[0];   b1.y = wp1[sN];
        c00 = WMMA_F32(a0, b0, c00);
        c10 = WMMA_F32(a1, b0, c10);
        c01 = WMMA_F32(a0, b1, c01);
        c11 = WMMA_F32(a1, b1, c11);
        ap0 += 4; ap1 += 4;
        wp0 += 4 * sN; wp1 += 4 * sN;
    }

    float* out0 = P + (size_t)blockIdx.y * 32 * sN + n0 + lo;
    float* out1 = out0 + 16;
#pragma unroll
    for (int r = 0; r < 8; ++r) {
        const size_t rm0 = (size_t)(r + 8 * hi) * sN;
        const size_t rm1 = (size_t)(16 + r + 8 * hi) * sN;
        out0[rm0] = c00[r];
        out0[rm1] = c10[r];
        out1[rm0] = c01[r];
        out1[rm1] = c11[r];
    }
}

// --------------------------- split-K epilogues -----------------------------
__global__ void reduce_sum_k(const float* __restrict__ P,
                             float* __restrict__ out, int total) {
    int i = blockIdx.x * 256 + threadIdx.x;
    if (i >= total) return;
    float a = 0.f;
    for (int s = 0; s < SPLITK; ++s) a += P[(size_t)s * total + i];
    out[i] = a;
}

__global__ void reduce_add_k(const float* __restrict__ P,
                             const float* __restrict__ res,
                             float* __restrict__ out, int total) {
    int i = blockIdx.x * 256 + threadIdx.x;
    if (i >= total) return;
    float a = 0.f;
    for (int s = 0; s < SPLITK; ++s) a += P[(size_t)s * total + i];
    out[i] = a + res[i];
}

__global__ void swiglu_k(const float* __restrict__ P1,
                         const float* __restrict__ P3,
                         float* __restrict__ out, int total) {
    int i = blockIdx.x * 256 + threadIdx.x;
    if (i >= total) return;
    float a = 0.f, bb = 0.f;
    for (int s = 0; s < SPLITK; ++s) {
        a  += P1[(size_t)s * total + i];
        bb += P3[(size_t)s * total + i];
    }
    out[i] = (a / (1.f + __expf(-a))) * bb;   // silu(a) * b
}

// -------------------------------- RoPE -------------------------------------
__global__ void rope_k(float* __restrict__ q, float* __restrict__ k,
                       const float* __restrict__ cs, const float* __restrict__ sn) {
    const int TOT = BATCH * (NHQ + NHKV) * (HD / 2);
    int t = blockIdx.x * 256 + threadIdx.x;
    if (t >= TOT) return;
    int i  = t & 63;
    int hh = (t >> 6) % (NHQ + NHKV);
    int b  = t / ((NHQ + NHKV) * 64);
    float c = cs[i], s = sn[i];
    float* p = (hh < NHQ) ? q + ((size_t)(b * NHQ + hh) * HD + 2 * i)
                          : k + ((size_t)(b * NHKV + (hh - NHQ)) * HD + 2 * i);
    float x0 = p[0], x1 = p[1];
    p[0] = x0 * c - x1 * s;
    p[1] = x0 * s + x1 * c;
}

// ------------------------- fused GQA attention -----------------------------
// One block per (batch, kv head): 4 Q heads share every K/V row read.
// Cache row at position L-1 is replaced by the freshly computed k/v
// (reference semantics without mutating the input cache).
__global__ __launch_bounds__(256) void attn_k(const float* __restrict__ q,
                                              const float* __restrict__ knew,
                                              const float* __restrict__ vnew,
                                              const float* __restrict__ ck,
                                              const float* __restrict__ cv,
                                              const int* __restrict__ sp,
                                              float* __restrict__ out) {
    __shared__ float sc[4][MAXSEQ + 32];
    __shared__ float qs[4 * HD];
    __shared__ float red[256];
    const int tid = threadIdx.x;
    const int bb  = blockIdx.x >> 3;
    const int g   = blockIdx.x & 7;
    const int L   = sp[0] + 1;

    for (int i = tid; i < 4 * HD; i += 256)
        qs[i] = q[((size_t)bb * NHQ + g * 4) * HD + i];
    __syncthreads();

    // scores: wave-parallel rows, lane-parallel dot + shuffle reduce
    const int wave = tid >> 5, lane = tid & 31;
    const float scale = 0.08838834764831845f;  // 1/sqrt(128)
    for (int l = wave; l < L; l += 8) {
        const float* krow = (l == L - 1)
            ? knew + ((size_t)bb * NHKV + g) * HD
            : ck + (((size_t)bb * MAXSEQ + l) * NHKV + g) * HD;
        float s0 = 0, s1 = 0, s2 = 0, s3 = 0;
#pragma unroll
        for (int j = 0; j < 4; ++j) {
            int d = lane + 32 * j;
            float kv = krow[d];
            s0 += qs[d] * kv;
            s1 += qs[HD + d] * kv;
            s2 += qs[2 * HD + d] * kv;
            s3 += qs[3 * HD + d] * kv;
        }
        for (int off = 16; off > 0; off >>= 1) {
            s0 += __shfl_xor(s0, off, 32);
            s1 += __shfl_xor(s1, off, 32);
            s2 += __shfl_xor(s2, off, 32);
            s3 += __shfl_xor(s3, off, 32);
        }
        if (lane == 0) {
            sc[0][l] = s0 * scale; sc[1][l] = s1 * scale;
            sc[2][l] = s2 * scale; sc[3][l] = s3 * scale;
        }
    }
    __syncthreads();

    // softmax per head
    for (int h = 0; h < 4; ++h) {
        float m = -1e30f;
        for (int l = tid; l < L; l += 256) m = fmaxf(m, sc[h][l]);
        red[tid] = m; __syncthreads();
        for (int o = 128; o > 0; o >>= 1) {
            if (tid < o) red[tid] = fmaxf(red[tid], red[tid + o]);
            __syncthreads();
        }
        m = red[0]; __syncthreads();
        float sum = 0.f;
        for (int l = tid; l < L; l += 256) {
            float e = __expf(sc[h][l] - m); sc[h][l] = e; sum += e;
        }
        red[tid] = sum; __syncthreads();
        for (int o = 128; o > 0; o >>= 1) {
            if (tid < o) red[tid] += red[tid + o];
            __syncthreads();
        }
        const float inv = 1.f / red[0]; __syncthreads();
        for (int l = tid; l < L; l += 256) sc[h][l] *= inv;
        __syncthreads();
    }

    // attn @ V: lane-per-channel (coalesced V rows), 2-way l split
    const int d    = tid & (HD - 1);
    const int half = tid >> 7;
    float a0 = 0, a1 = 0, a2 = 0, a3 = 0;
    for (int l = half; l < L; l += 2) {
        const float* vrow = (l == L - 1)
            ? vnew + ((size_t)bb * NHKV + g) * HD
            : cv + (((size_t)bb * MAXSEQ + l) * NHKV + g) * HD;
        float vv = vrow[d];
        a0 += sc[0][l] * vv; a1 += sc[1][l] * vv;
        a2 += sc[2][l] * vv; a3 += sc[3][l] * vv;
    }
    __syncthreads();
    float* tmp = &sc[0][0];
    tmp[(half * 4 + 0) * HD + d] = a0;
    tmp[(half * 4 + 1) * HD + d] = a1;
    tmp[(half * 4 + 2) * HD + d] = a2;
    tmp[(half * 4 + 3) * HD + d] = a3;
    __syncthreads();
    if (half == 0) {
#pragma unroll
        for (int h = 0; h < 4; ++h)
            out[(size_t)bb * DIM + (g * 4 + h) * HD + d] =
                tmp[h * HD + d] + tmp[(4 + h) * HD + d];
    }
}

// ------------------------------- driver ------------------------------------
extern "C" void kernel_launch(void* const* d_in, const int* in_sizes, int n_in,
                              void* d_out, int out_size, void* d_ws, size_t ws_size,
                              hipStream_t stream) {
    (void)in_sizes; (void)n_in; (void)out_size; (void)ws_size;
    const float* x   = (const float*)d_in[0];
    const int*   sp  = (const int*)d_in[1];
    const float* fc  = (const float*)d_in[2];
    const float* fs  = (const float*)d_in[3];
    const float* ck  = (const float*)d_in[4];
    const float* cv  = (const float*)d_in[5];
    const float* wq  = (const float*)d_in[6];
    const float* wk  = (const float*)d_in[7];
    const float* wv  = (const float*)d_in[8];
    const float* wo  = (const float*)d_in[9];
    const float* w1  = (const float*)d_in[10];
    const float* w2  = (const float*)d_in[11];
    const float* w3  = (const float*)d_in[12];
    const float* anw = (const float*)d_in[13];
    const float* fnw = (const float*)d_in[14];
    float* out = (float*)d_out;

    float* ws  = (float*)d_ws;
    float* h    = ws; ws += (size_t)BATCH * DIM;
    float* qb   = ws; ws += (size_t)BATCH * DIM;
    float* kbf  = ws; ws += (size_t)BATCH * NHKV * HD;
    float* vbf  = ws; ws += (size_t)BATCH * NHKV * HD;
    float* ao   = ws; ws += (size_t)BATCH * DIM;
    float* o1   = ws; ws += (size_t)BATCH * DIM;
    float* h2   = ws; ws += (size_t)BATCH * DIM;
    float* gb   = ws; ws += (size_t)BATCH * HIDDEN;
    float* pA   = ws; ws += (size_t)SPLITK * BATCH * HIDDEN;
    float* pB   = ws; ws += (size_t)SPLITK * BATCH * HIDDEN;

    const int TOT_D = BATCH * DIM;        // 131072
    const int TOT_K = BATCH * NHKV * HD;  // 32768
    const int TOT_H = BATCH * HIDDEN;     // 352256

    // attention block
    rmsnorm_k<<<BATCH, 256, 0, stream>>>(x, anw, h);
    gemm32_wmma<<<dim3(DIM / 256, SPLITK), 256, 0, stream>>>(h, wq, pA, DIM, DIM);
    reduce_sum_k<<<(TOT_D + 255) / 256, 256, 0, stream>>>(pA, qb, TOT_D);
    gemm32_wmma<<<dim3((NHKV * HD) / 256, SPLITK), 256, 0, stream>>>(h, wk, pA, DIM, NHKV * HD);
    reduce_sum_k<<<(TOT_K + 255) / 256, 256, 0, stream>>>(pA, kbf, TOT_K);
    gemm32_wmma<<<dim3((NHKV * HD) / 256, SPLITK), 256, 0, stream>>>(h, wv, pA, DIM, NHKV * HD);
    reduce_sum_k<<<(TOT_K + 255) / 256, 256, 0, stream>>>(pA, vbf, TOT_K);
    rope_k<<<(BATCH * (NHQ + NHKV) * (HD / 2) + 255) / 256, 256, 0, stream>>>(qb, kbf, fc, fs);
    attn_k<<<BATCH * NHKV, 256, 0, stream>>>(qb, kbf, vbf, ck, cv, sp, ao);
    gemm32_wmma<<<dim3(DIM / 256, SPLITK), 256, 0, stream>>>(ao, wo, pA, DIM, DIM);
    reduce_add_k<<<(TOT_D + 255) / 256, 256, 0, stream>>>(pA, x, o1, TOT_D);

    // FFN block
    rmsnorm_k<<<BATCH, 256, 0, stream>>>(o1, fnw, h2);
    gemm32_wmma<<<dim3(HIDDEN / 256, SPLITK), 256, 0, stream>>>(h2, w1, pA, DIM, HIDDEN);
    gemm32_wmma<<<dim3(HIDDEN / 256, SPLITK), 256, 0, stream>>>(h2, w3, pB, DIM, HIDDEN);
    swiglu_k<<<(TOT_H + 255) / 256, 256, 0, stream>>>(pA, pB, gb, TOT_H);
    gemm32_wmma<<<dim3(DIM / 256, SPLITK), 256, 0, stream>>>(gb, w2, pA, HIDDEN, DIM);
    reduce_add_k<<<(TOT_D + 255) / 256, 256, 0, stream>>>(pA, o1, out, TOT_D);
}